// OctantQuery_36687610643110
// MI455X (gfx1250) — compile-verified
//
#include <hip/hip_runtime.h>
#include <stdint.h>

// Octant radius-ball query for pcs[16][3][200000] (f32) -> int32 out[16][8][512].
// One workgroup per (batch, octant). Double-buffered async global->LDS staging
// (CDNA5 GLOBAL_LOAD_ASYNC_TO_LDS_B128 + s_wait_asynccnt), wave32 ballot scans,
// uniform early exit once 512 samples are found.

#define NPTS     200000
#define NBATCH   16
#define MAXS     512
#define TPB      256
#define PPT      4                       // points per thread (float4 / B128)
#define CHUNK    (TPB * PPT)             // 1024 points per chunk
#define NCHUNKS  ((NPTS + CHUNK - 1) / CHUNK)   // 196
#define NWAVES   (TPB / 32)

#define AS1 __attribute__((address_space(1)))
#define AS3 __attribute__((address_space(3)))

typedef int v4i __attribute__((ext_vector_type(4)));

__device__ __forceinline__ void async_cp16(const float* gsrc, float* ldst) {
#if __has_builtin(__builtin_amdgcn_global_load_async_to_lds_b128)
  __builtin_amdgcn_global_load_async_to_lds_b128((v4i*)gsrc, (AS3 v4i*)ldst, 0, 0);
#else
  unsigned loff = (unsigned)(size_t)(AS3 void*)ldst;   // 32-bit LDS byte offset
  asm volatile("global_load_async_to_lds_b128 %0, %1, off"
               :: "v"(loff), "v"(gsrc) : "memory");
#endif
}

__device__ __forceinline__ void wait_async0() {
#if __has_builtin(__builtin_amdgcn_s_wait_asynccnt)
  __builtin_amdgcn_s_wait_asynccnt(0);
  asm volatile("" ::: "memory");          // compiler-level ordering fence
#else
  asm volatile("s_wait_asynccnt 0x0" ::: "memory");
#endif
}

__device__ __forceinline__ unsigned ballot32(bool p) {
#if __has_builtin(__builtin_amdgcn_ballot_w32)
  return __builtin_amdgcn_ballot_w32(p);
#else
  return (unsigned)__ballot(p);
#endif
}

__global__ __launch_bounds__(TPB) void OctantQuery_36687610643110_kernel(
    const float* __restrict__ pcs, int* __restrict__ out) {
  const int o    = (int)blockIdx.x;       // octant 0..7
  const int b    = (int)blockIdx.y;       // batch 0..15
  const int tid  = (int)threadIdx.x;
  const int lane = tid & 31;
  const int wv   = tid >> 5;
  const unsigned ltm = (1u << lane) - 1u;

  const float* px = pcs + ((size_t)(b * 3 + 0)) * NPTS;
  const float* py = pcs + ((size_t)(b * 3 + 1)) * NPTS;
  const float* pz = pcs + ((size_t)(b * 3 + 2)) * NPTS;
  int* outp = out + ((size_t)(b * 8 + o)) * MAXS;

  __shared__ int wsum[NWAVES];
  __shared__ __align__(16) float sx[2][CHUNK];
  __shared__ __align__(16) float sy[2][CHUNK];
  __shared__ __align__(16) float sz[2][CHUNK];

  const int myoff = tid * PPT;

  // Prologue: stage chunk 0 (always fully in range: CHUNK <= NPTS).
  async_cp16(px + myoff, &sx[0][myoff]);
  async_cp16(py + myoff, &sy[0][myoff]);
  async_cp16(pz + myoff, &sz[0][myoff]);

  int running = 0;   // samples emitted so far for this (b, o) — uniform
  int cur = 0;

  for (int c = 0; c < NCHUNKS; ++c) {
    const int gidx0 = c * CHUNK + myoff;
    const bool act  = gidx0 < NPTS;       // 4-aligned tail: whole float4 valid or not

    wait_async0();                        // chunk c resident in buffers[cur]

    // Kick off chunk c+1 into the other buffer (overlaps this chunk's compute).
    if (c + 1 < NCHUNKS) {
      const int ng = gidx0 + CHUNK;
      if (ng < NPTS) {
        const int nb = cur ^ 1;
        async_cp16(px + ng, &sx[nb][myoff]);
        async_cp16(py + ng, &sy[nb][myoff]);
        async_cp16(pz + ng, &sz[nb][myoff]);
      }
    }

    // Each thread consumes exactly the LDS bytes it staged (wave-local dep).
    const float4 xv = *(const float4*)&sx[cur][myoff];
    const float4 yv = *(const float4*)&sy[cur][myoff];
    const float4 zv = *(const float4*)&sz[cur][myoff];

    bool m[PPT];
    {
      const float X[PPT] = {xv.x, xv.y, xv.z, xv.w};
      const float Y[PPT] = {yv.x, yv.y, yv.z, yv.w};
      const float Z[PPT] = {zv.x, zv.y, zv.z, zv.w};
#pragma unroll
      for (int j = 0; j < PPT; ++j) {
        const float x = X[j], y = Y[j], z = Z[j];
        // No FMA contraction: match reference's mul/mul/mul/add/add rounding.
        const float s = __fadd_rn(__fadd_rn(__fmul_rn(x, x), __fmul_rn(y, y)),
                                  __fmul_rn(z, z));
        const int oc = ((x >= 0.0f) ? 4 : 0) + ((y >= 0.0f) ? 2 : 0) +
                       ((z >= 0.0f) ? 1 : 0);
        m[j] = act && (s <= 1.0f) && (oc == o);
      }
    }

    // Wave32 ordered scan: chunk order is base + 4*lane + j (lane-major).
    unsigned bl[PPT];
#pragma unroll
    for (int j = 0; j < PPT; ++j) bl[j] = ballot32(m[j]);

    int waveCnt = 0, lanePre = 0;
#pragma unroll
    for (int j = 0; j < PPT; ++j) {
      waveCnt += __popc(bl[j]);
      lanePre += __popc(bl[j] & ltm);
    }

    __syncthreads();                      // protect wsum from previous round
    if (lane == 0) wsum[wv] = waveCnt;
    __syncthreads();

    int waveBase = 0, chunkTot = 0;
#pragma unroll
    for (int w = 0; w < NWAVES; ++w) {
      const int v = wsum[w];
      chunkTot += v;
      waveBase += (w < wv) ? v : 0;
    }

    int r = running + waveBase + lanePre;
#pragma unroll
    for (int j = 0; j < PPT; ++j) {
      if (m[j]) {
        if (r < MAXS) outp[r] = gidx0 + j;
        ++r;
      }
    }

    running += chunkTot;                  // uniform across the block
    if (running >= MAXS) break;           // uniform early exit (~21k pts scanned)
    cur ^= 1;
  }

  // Pad unused slots with DEFAULT = -1 (full d_out rewrite every call).
  const int cnt = running < MAXS ? running : MAXS;
  for (int i = cnt + tid; i < MAXS; i += TPB) outp[i] = -1;
}

extern "C" void kernel_launch(void* const* d_in, const int* in_sizes, int n_in,
                              void* d_out, int out_size, void* d_ws, size_t ws_size,
                              hipStream_t stream) {
  (void)in_sizes; (void)n_in; (void)out_size; (void)d_ws; (void)ws_size;
  const float* pcs = (const float*)d_in[0];
  int* out = (int*)d_out;
  dim3 grid(8, NBATCH, 1);
  OctantQuery_36687610643110_kernel<<<grid, TPB, 0, stream>>>(pcs, out);
}